// EGNN_22574348108108
// MI455X (gfx1250) — compile-verified
//
#include <hip/hip_runtime.h>
#include <hip/hip_bf16.h>

typedef __attribute__((ext_vector_type(2))) float v2f;
typedef __attribute__((ext_vector_type(8))) float v8f;

#define DIM 128
#define BN_EPS 1e-5f

// ---------------------------------------------------------------------------
// AtomEncoder: h[n,d] = sum_f atom_emb[f, x[n,f], d]
// ---------------------------------------------------------------------------
__global__ __launch_bounds__(256) void atom_encoder_kernel(
    const int* __restrict__ x, const float* __restrict__ atom_emb,
    float* __restrict__ h, int N) {
  int idx = blockIdx.x * 256 + threadIdx.x;
  if (idx >= N * DIM) return;
  int n = idx >> 7;
  int d = idx & (DIM - 1);
  float s = 0.0f;
#pragma unroll
  for (int f = 0; f < 9; ++f) {
    int v = x[n * 9 + f];  // 0..63
    s += atom_emb[((size_t)f * 64 + v) * DIM + d];
  }
  h[idx] = s;
}

// ---------------------------------------------------------------------------
// degree: deg[row[e]] += 1
// ---------------------------------------------------------------------------
__global__ __launch_bounds__(256) void degree_kernel(
    const int* __restrict__ row, float* __restrict__ deg, int E) {
  int e = blockIdx.x * 256 + threadIdx.x;
  if (e < E) atomicAdd(&deg[row[e]], 1.0f);
}

__global__ __launch_bounds__(256) void dinv_kernel(
    float* __restrict__ deg, float* __restrict__ dinv, int N) {
  int n = blockIdx.x * 256 + threadIdx.x;
  if (n >= N) return;
  float dd = deg[n] + 1.0f;
  deg[n] = dd;
  dinv[n] = rsqrtf(dd);
}

// ---------------------------------------------------------------------------
// Linear: hl = h @ W + b   via V_WMMA_F32_16X16X4_F32
// Block: 256 threads = 8 waves. Block computes a 16x128 row panel.
// Wave w computes the 16x16 tile at columns [16w, 16w+16).
// A (16x4 f32): lane<16 holds A[m=lane][k,k+1], lane>=16 A[m=lane-16][k+2,k+3]
// B (4x16 f32): lane<16 holds B[k][n=lane],B[k+1][n]; lane>=16 B[k+2][n-16],B[k+3][n-16]
// C/D (16x16 f32): vgpr r: lanes 0-15 -> M=r, lanes 16-31 -> M=r+8
// ---------------------------------------------------------------------------
#define A_PITCH 132  // pad to avoid LDS bank conflicts on stride-row reads

__global__ __launch_bounds__(256) void linear_wmma_kernel(
    const float* __restrict__ h, const float* __restrict__ W,
    const float* __restrict__ bias, float* __restrict__ hl, int N) {
  __shared__ float Atile[16 * A_PITCH];

  int tid = threadIdx.x;
  int rowBase = blockIdx.x * 16;
  if (rowBase >= N) return;

  // Stage 16x128 panel of h into LDS (coalesced).
  for (int i = tid; i < 16 * DIM; i += 256) {
    int r = i >> 7;
    int k = i & (DIM - 1);
    Atile[r * A_PITCH + k] = h[(size_t)(rowBase + r) * DIM + k];
  }
  __syncthreads();

  int wave = tid >> 5;        // column tile 0..7
  int lane = tid & 31;
  int hv = lane >> 4;         // half: 0 or 1
  int lid = lane & 15;
  int colBase = wave * 16;

  v8f acc = {};
  const float* srcA = &Atile[lid * A_PITCH];
#pragma unroll
  for (int k = 0; k < DIM; k += 4) {
    int ka = k + 2 * hv;
    v2f a, bf;
    a.x = srcA[ka];
    a.y = srcA[ka + 1];
    bf.x = W[(size_t)ka * DIM + colBase + lid];
    bf.y = W[(size_t)(ka + 1) * DIM + colBase + lid];
    acc = __builtin_amdgcn_wmma_f32_16x16x4_f32(
        false, a, false, bf, (short)0, acc, false, false);
  }

  int col = colBase + lid;
  float bb = bias[col];
#pragma unroll
  for (int r = 0; r < 8; ++r) {
    int rowm = rowBase + r + 8 * hv;
    hl[(size_t)rowm * DIM + col] = acc[r] + bb;
  }
}

// ---------------------------------------------------------------------------
// Edge message pass: aggr[col] += norm * relu(hl[row] + bond_embed(edge))
// One wave (32 lanes) per edge, float4 per lane (128 channels).
// ---------------------------------------------------------------------------
__global__ __launch_bounds__(256) void edge_kernel(
    const int* __restrict__ row, const int* __restrict__ col,
    const int* __restrict__ edge_attr, const float* __restrict__ bond_emb_l,
    const float* __restrict__ hl, const float* __restrict__ dinv,
    float* __restrict__ aggr, int E) {
  int e = blockIdx.x * 8 + (threadIdx.x >> 5);
  if (e >= E) return;
  int lane = threadIdx.x & 31;

  int r = row[e];
  int c = col[e];
  float nrm = dinv[r] * dinv[c];
  int a0 = edge_attr[e * 3 + 0];
  int a1 = edge_attr[e * 3 + 1];
  int a2 = edge_attr[e * 3 + 2];

  const float4* t0 = (const float4*)(bond_emb_l + (size_t)(0 * 8 + a0) * DIM);
  const float4* t1 = (const float4*)(bond_emb_l + (size_t)(1 * 8 + a1) * DIM);
  const float4* t2 = (const float4*)(bond_emb_l + (size_t)(2 * 8 + a2) * DIM);
  const float4* hr = (const float4*)(hl + (size_t)r * DIM);
  float* ag = aggr + (size_t)c * DIM;

  float4 vh = hr[lane];
  float4 v0 = t0[lane];
  float4 v1 = t1[lane];
  float4 v2 = t2[lane];
  float m0 = fmaxf(vh.x + v0.x + v1.x + v2.x, 0.0f) * nrm;
  float m1 = fmaxf(vh.y + v0.y + v1.y + v2.y, 0.0f) * nrm;
  float m2 = fmaxf(vh.z + v0.z + v1.z + v2.z, 0.0f) * nrm;
  float m3 = fmaxf(vh.w + v0.w + v1.w + v2.w, 0.0f) * nrm;
  int d = lane * 4;
  atomicAdd(&ag[d + 0], m0);
  atomicAdd(&ag[d + 1], m1);
  atomicAdd(&ag[d + 2], m2);
  atomicAdd(&ag[d + 3], m3);
}

// ---------------------------------------------------------------------------
// Combine: h = aggr + relu(hl + root)/deg; accumulate BN stats per channel.
// 128 threads (one per channel), each block walks 256 nodes.
// ---------------------------------------------------------------------------
__global__ __launch_bounds__(128) void combine_kernel(
    const float* __restrict__ aggr, const float* __restrict__ hl,
    const float* __restrict__ root_l, const float* __restrict__ deg,
    float* __restrict__ h, float* __restrict__ ssum, float* __restrict__ ssq,
    int N, int doStats) {
  int d = threadIdx.x;
  int n0 = blockIdx.x * 256;
  float rt = root_l[d];
  float s = 0.0f, sq = 0.0f;
  for (int i = 0; i < 256; ++i) {
    int n = n0 + i;
    if (n >= N) break;
    size_t off = (size_t)n * DIM + d;
    float v = aggr[off] + fmaxf(hl[off] + rt, 0.0f) / deg[n];
    h[off] = v;
    s += v;
    sq += v * v;
  }
  if (doStats) {
    atomicAdd(&ssum[d], s);
    atomicAdd(&ssq[d], sq);
  }
}

__global__ __launch_bounds__(128) void stats_finalize_kernel(
    const float* __restrict__ ssum, const float* __restrict__ ssq,
    float* __restrict__ mean, float* __restrict__ istd, int N) {
  int d = threadIdx.x;
  float invN = 1.0f / (float)N;
  float mu = ssum[d] * invN;
  float var = ssq[d] * invN - mu * mu;
  mean[d] = mu;
  istd[d] = rsqrtf(var + BN_EPS);
}

__global__ __launch_bounds__(256) void bn_relu_kernel(
    float* __restrict__ h, const float* __restrict__ mean,
    const float* __restrict__ istd, const float* __restrict__ gamma_l,
    const float* __restrict__ beta_l, int N) {
  int idx = blockIdx.x * 256 + threadIdx.x;
  if (idx >= N * DIM) return;
  int d = idx & (DIM - 1);
  float v = (h[idx] - mean[d]) * istd[d] * gamma_l[d] + beta_l[d];
  h[idx] = fmaxf(v, 0.0f);
}

// ---------------------------------------------------------------------------
// Global mean pool (sum + count via atomics)
// ---------------------------------------------------------------------------
__global__ __launch_bounds__(256) void pool_kernel(
    const float* __restrict__ h, const int* __restrict__ batch,
    float* __restrict__ hg, float* __restrict__ counts, int N) {
  int idx = blockIdx.x * 256 + threadIdx.x;
  if (idx >= N * DIM) return;
  int n = idx >> 7;
  int d = idx & (DIM - 1);
  int g = batch[n];
  atomicAdd(&hg[(size_t)g * DIM + d], h[idx]);
  if (d == 0) atomicAdd(&counts[g], 1.0f);
}

// ---------------------------------------------------------------------------
// Output MLP: relu(mean) -> relu(@W1+b1) -> @W2+b2. One block per graph.
// ---------------------------------------------------------------------------
__global__ __launch_bounds__(128) void mlp_kernel(
    const float* __restrict__ hg, const float* __restrict__ counts,
    const float* __restrict__ W1, const float* __restrict__ b1,
    const float* __restrict__ W2, const float* __restrict__ b2,
    float* __restrict__ out, int T) {
  __shared__ float sh[DIM];
  __shared__ float st[DIM];
  int g = blockIdx.x;
  int d = threadIdx.x;
  float cnt = fmaxf(counts[g], 1.0f);
  sh[d] = fmaxf(hg[(size_t)g * DIM + d] / cnt, 0.0f);
  __syncthreads();
  float acc = b1[d];
  for (int k = 0; k < DIM; ++k) acc += sh[k] * W1[k * DIM + d];
  st[d] = fmaxf(acc, 0.0f);
  __syncthreads();
  if (d < T) {
    float o = b2[d];
    for (int k = 0; k < DIM; ++k) o += st[k] * W2[k * T + d];
    out[g * T + d] = o;
  }
}

// ---------------------------------------------------------------------------
// Host launcher
// ---------------------------------------------------------------------------
extern "C" void kernel_launch(void* const* d_in, const int* in_sizes, int n_in,
                              void* d_out, int out_size, void* d_ws, size_t ws_size,
                              hipStream_t stream) {
  const int N = in_sizes[0] / 9;       // 100000
  const int E = in_sizes[1] / 2;       // 800000
  const int G = 512;
  const int T = 10;
  const int L = 4;

  const int*   x         = (const int*)d_in[0];
  const int*   edge_idx  = (const int*)d_in[1];
  const int*   batch     = (const int*)d_in[2];
  const int*   edge_attr = (const int*)d_in[3];
  const float* atom_emb  = (const float*)d_in[4];
  const float* W         = (const float*)d_in[5];
  const float* b         = (const float*)d_in[6];
  const float* root      = (const float*)d_in[7];
  const float* bond_emb  = (const float*)d_in[8];
  const float* gamma     = (const float*)d_in[9];
  const float* beta      = (const float*)d_in[10];
  const float* W1        = (const float*)d_in[11];
  const float* b1        = (const float*)d_in[12];
  const float* W2        = (const float*)d_in[13];
  const float* b2        = (const float*)d_in[14];
  float* out = (float*)d_out;

  const int* e_row = edge_idx;          // source j
  const int* e_col = edge_idx + E;      // target i

  // Workspace carve (256B aligned)
  char* ws = (char*)d_ws;
  size_t off = 0;
  auto alloc = [&](size_t bytes) {
    char* p = ws + off;
    off += (bytes + 255) & ~(size_t)255;
    return p;
  };
  float* h    = (float*)alloc((size_t)N * DIM * sizeof(float));
  float* hl   = (float*)alloc((size_t)N * DIM * sizeof(float));
  float* aggr = (float*)alloc((size_t)N * DIM * sizeof(float));
  float* deg  = (float*)alloc((size_t)N * sizeof(float));
  float* dinv = (float*)alloc((size_t)N * sizeof(float));
  float* ssum = (float*)alloc(DIM * sizeof(float));
  float* ssq  = (float*)alloc(DIM * sizeof(float));
  float* mean = (float*)alloc(DIM * sizeof(float));
  float* istd = (float*)alloc(DIM * sizeof(float));
  float* hg   = (float*)alloc((size_t)G * DIM * sizeof(float));
  float* cnts = (float*)alloc((size_t)G * sizeof(float));
  (void)ws_size;

  const int nd_blocks = (N * DIM + 255) / 256;

  // AtomEncoder + degree
  hipMemsetAsync(deg, 0, (size_t)N * sizeof(float), stream);
  atom_encoder_kernel<<<nd_blocks, 256, 0, stream>>>(x, atom_emb, h, N);
  degree_kernel<<<(E + 255) / 256, 256, 0, stream>>>(e_row, deg, E);
  dinv_kernel<<<(N + 255) / 256, 256, 0, stream>>>(deg, dinv, N);

  for (int l = 0; l < L; ++l) {
    const float* Wl    = W + (size_t)l * DIM * DIM;
    const float* bl    = b + (size_t)l * DIM;
    const float* rootl = root + (size_t)l * DIM;
    const float* bemb  = bond_emb + (size_t)l * 3 * 8 * DIM;
    const int doStats = (l < L - 1) ? 1 : 0;

    linear_wmma_kernel<<<(N + 15) / 16, 256, 0, stream>>>(h, Wl, bl, hl, N);

    hipMemsetAsync(aggr, 0, (size_t)N * DIM * sizeof(float), stream);
    hipMemsetAsync(ssum, 0, DIM * sizeof(float), stream);
    hipMemsetAsync(ssq, 0, DIM * sizeof(float), stream);

    edge_kernel<<<(E + 7) / 8, 256, 0, stream>>>(
        e_row, e_col, edge_attr, bemb, hl, dinv, aggr, E);

    combine_kernel<<<(N + 255) / 256, 128, 0, stream>>>(
        aggr, hl, rootl, deg, h, ssum, ssq, N, doStats);

    if (doStats) {
      stats_finalize_kernel<<<1, 128, 0, stream>>>(ssum, ssq, mean, istd, N);
      bn_relu_kernel<<<nd_blocks, 256, 0, stream>>>(
          h, mean, istd, gamma + (size_t)l * DIM, beta + (size_t)l * DIM, N);
    }
  }

  // Pool + MLP
  hipMemsetAsync(hg, 0, (size_t)G * DIM * sizeof(float), stream);
  hipMemsetAsync(cnts, 0, (size_t)G * sizeof(float), stream);
  pool_kernel<<<nd_blocks, 256, 0, stream>>>(h, batch, hg, cnts, N);
  mlp_kernel<<<G, 128, 0, stream>>>(hg, cnts, W1, b1, W2, b2, out, T);
  (void)out_size; (void)n_in;
}